// ColorCNN_31894427140213
// MI455X (gfx1250) — compile-verified
//
#include <hip/hip_runtime.h>
#include <hip/hip_bf16.h>

typedef __attribute__((ext_vector_type(2))) float v2f;
typedef __attribute__((ext_vector_type(4))) float v4f;
typedef __attribute__((ext_vector_type(8))) float v8f;

#define HW 65536            // 256*256
#define NPIX (8 * HW)       // 524288 pixels total
#define NC 64               // num_colors

// ---------------------------------------------------------------------------
// K1: x = feat[p,0:64] . w_bneck^T + b  (WMMA f32 16x16x4), store x pixel-major,
//     accumulate per-channel sum / sumsq for BatchNorm training stats.
// ---------------------------------------------------------------------------
__global__ __launch_bounds__(256) void k1_bneck(const float* __restrict__ feat,
                                                const float* __restrict__ wb,
                                                const float* __restrict__ bb,
                                                float* __restrict__ xws,
                                                float* __restrict__ stats /*[32]*/) {
  __shared__ float red[32];
  const int lane = threadIdx.x & 31;
  const int wid  = threadIdx.x >> 5;
  const int g    = lane >> 4;     // K-half selector
  const int n    = lane & 15;     // N (and M) index within tile

  // B operand (w_bneck, K x N with B[k][n] = w_bneck[n][c0+k]) for all 16 k-steps
  v2f Bk[16];
#pragma unroll
  for (int kk = 0; kk < 16; ++kk) {
    int c = 4 * kk + 2 * g;
    Bk[kk].x = wb[n * 64 + c];
    Bk[kk].y = wb[n * 64 + c + 1];
  }
  const float bias = bb[n];

  float s = 0.f, ss = 0.f;
  const int TPW = 4;  // 16-pixel tiles per wave
  const int tile0 = (blockIdx.x * 8 + wid) * TPW;

  for (int t = 0; t < TPW; ++t) {
    const int pbase = (tile0 + t) * 16;
    const int b  = pbase >> 16;
    const int hw = pbase & (HW - 1);
    const float* fb = feat + (size_t)b * (64 * HW) + hw + n;  // this lane's pixel
    v8f acc = {};
#pragma unroll
    for (int kk = 0; kk < 16; ++kk) {
      const int c = 4 * kk + 2 * g;
      v2f a;
      a.x = fb[(size_t)c * HW];
      a.y = fb[(size_t)(c + 1) * HW];
      acc = __builtin_amdgcn_wmma_f32_16x16x4_f32(false, a, false, Bk[kk],
                                                  (short)0, acc, false, false);
    }
    // acc[r] = x[pbase + r + 8g][n]
#pragma unroll
    for (int r = 0; r < 8; ++r) {
      const float xv = acc[r] + bias;
      s += xv;
      ss += xv * xv;
      xws[(size_t)(pbase + r + 8 * g) * 16 + n] = xv;
    }
  }
  if (threadIdx.x < 32) red[threadIdx.x] = 0.f;
  __syncthreads();
  atomicAdd(&red[n], s);
  atomicAdd(&red[16 + n], ss);
  __syncthreads();
  if (threadIdx.x < 32) atomicAdd(&stats[threadIdx.x], red[threadIdx.x]);
}

// ---------------------------------------------------------------------------
// K3: BN(scale/shift)+ReLU on x, WMMA GEMM2 -> 64 color logits, per-pixel
//     top-4 + softmax, sparse-scatter m, accumulate num/den, store compact.
// ---------------------------------------------------------------------------
__global__ __launch_bounds__(128) void k3_main(const float* __restrict__ xws,
                                               const float* __restrict__ wmask,
                                               const float* __restrict__ gamma,
                                               const float* __restrict__ beta,
                                               const float* __restrict__ stats,
                                               const float* __restrict__ img,
                                               float* __restrict__ m_out,
                                               float* __restrict__ numacc,
                                               float* __restrict__ denacc,
                                               unsigned int* __restrict__ tkidx,
                                               float* __restrict__ tkw) {
  __shared__ float bn_scale[16], bn_shift[16];
  __shared__ float weff[64 * 16];
  __shared__ float numl[192];
  __shared__ float denl[64];
  __shared__ float mtile[4][32 * 65];   // per-wave tile, pitch 65 (odd => no bank conflicts)

  const int tid = threadIdx.x;
  if (tid < 16) {
    const float mu  = stats[tid] * (1.f / NPIX);
    const float var = stats[16 + tid] * (1.f / NPIX) - mu * mu;
    const float inv = rsqrtf(var + 1e-5f);
    const float sc  = gamma[tid] * inv;
    bn_scale[tid] = sc;
    bn_shift[tid] = beta[tid] - mu * sc;
  }
  // effective mask weight: mean over 4 groups of 64 output channels
  for (int e = tid; e < 1024; e += 128) {
    const int nn = e >> 4, c = e & 15;
    weff[e] = 0.25f * (wmask[nn * 16 + c] + wmask[(nn + 64) * 16 + c] +
                       wmask[(nn + 128) * 16 + c] + wmask[(nn + 192) * 16 + c]);
  }
  for (int e = tid; e < 192; e += 128) numl[e] = 0.f;
  if (tid < 64) denl[tid] = 0.f;
  __syncthreads();

  const int lane = tid & 31, wid = tid >> 5;
  const int g = lane >> 4, nl = lane & 15;

  // B operand: B2[nt][kk][k][n] = weff[nt*16+n][4kk+k]
  v2f B2[4][4];
#pragma unroll
  for (int nt = 0; nt < 4; ++nt)
#pragma unroll
    for (int kk = 0; kk < 4; ++kk) {
      const int c = 4 * kk + 2 * g;
      B2[nt][kk].x = weff[(nt * 16 + nl) * 16 + c];
      B2[nt][kk].y = weff[(nt * 16 + nl) * 16 + c + 1];
    }

  const int SEG = 8;                                   // 32-pixel segments per wave
  const int seg0 = (blockIdx.x * 4 + wid) * SEG;
  float* mt = &mtile[wid][0];

  for (int sgi = 0; sgi < SEG; ++sgi) {
    const int pseg = (seg0 + sgi) * 32;
    // two 16-pixel WMMA tiles -> LDS rows 0..31
    for (int half = 0; half < 2; ++half) {
      const int pbase = pseg + half * 16;
      const int pix = pbase + nl;
      v2f A[4];
#pragma unroll
      for (int kk = 0; kk < 4; ++kk) {
        const int c = 4 * kk + 2 * g;
        float ax = xws[(size_t)pix * 16 + c];
        float ay = xws[(size_t)pix * 16 + c + 1];
        ax = fmaxf(ax * bn_scale[c] + bn_shift[c], 0.f);
        ay = fmaxf(ay * bn_scale[c + 1] + bn_shift[c + 1], 0.f);
        A[kk].x = ax;
        A[kk].y = ay;
      }
#pragma unroll
      for (int nt = 0; nt < 4; ++nt) {
        v8f acc = {};
#pragma unroll
        for (int kk = 0; kk < 4; ++kk)
          acc = __builtin_amdgcn_wmma_f32_16x16x4_f32(false, A[kk], false, B2[nt][kk],
                                                      (short)0, acc, false, false);
#pragma unroll
        for (int r = 0; r < 8; ++r)
          mt[(half * 16 + r + 8 * g) * 65 + nt * 16 + nl] = acc[r];
      }
    }
    // per-lane pixel: serial top-4 over 64 logits (strict > matches jax tie-break)
    const int p = pseg + lane;
    const float* row = &mt[lane * 65];
    float v0 = -1e30f, v1 = -1e30f, v2 = -1e30f, v3 = -1e30f;
    int i0 = 0, i1 = 0, i2 = 0, i3 = 0;
    for (int j = 0; j < 64; ++j) {
      const float v = row[j];
      if (v > v0)      { v3=v2;i3=i2; v2=v1;i2=i1; v1=v0;i1=i0; v0=v;i0=j; }
      else if (v > v1) { v3=v2;i3=i2; v2=v1;i2=i1; v1=v;i1=j; }
      else if (v > v2) { v3=v2;i3=i2; v2=v;i2=j; }
      else if (v > v3) { v3=v;i3=j; }
    }
    const float e1 = __expf(v1 - v0), e2 = __expf(v2 - v0), e3 = __expf(v3 - v0);
    const float inv = 1.f / (1.f + e1 + e2 + e3);
    const float w0 = inv, w1 = e1 * inv, w2 = e2 * inv, w3 = e3 * inv;

    tkidx[p] = (unsigned)i0 | ((unsigned)i1 << 8) | ((unsigned)i2 << 16) | ((unsigned)i3 << 24);
    v4f wv; wv.x = w0; wv.y = w1; wv.z = w2; wv.w = w3;
    ((v4f*)tkw)[p] = wv;

    const int b = p >> 16;
    const int hw = p & (HW - 1);
    float* mo = m_out + (size_t)b * 64 * HW + hw;    // m pre-zeroed by memset
    mo[(size_t)i0 * HW] = w0;
    mo[(size_t)i1 * HW] = w1;
    mo[(size_t)i2 * HW] = w2;
    mo[(size_t)i3 * HW] = w3;

    atomicAdd(&denl[i0], w0); atomicAdd(&denl[i1], w1);
    atomicAdd(&denl[i2], w2); atomicAdd(&denl[i3], w3);
    const float* ib = img + (size_t)b * 3 * HW + hw;
    const float ir = ib[0], ig = ib[HW], ibv = ib[2 * HW];
#pragma unroll
    for (int k = 0; k < 4; ++k) {
      const int ii = (k == 0) ? i0 : (k == 1) ? i1 : (k == 2) ? i2 : i3;
      const float ww = (k == 0) ? w0 : (k == 1) ? w1 : (k == 2) ? w2 : w3;
      atomicAdd(&numl[ii],       ww * ir);
      atomicAdd(&numl[64 + ii],  ww * ig);
      atomicAdd(&numl[128 + ii], ww * ibv);
    }
  }
  __syncthreads();
  const int b = (blockIdx.x * 1024) >> 16;           // block covers 1024 px in one image
  if (tid < 64) atomicAdd(&denacc[b * 64 + tid], denl[tid]);
  for (int e = tid; e < 192; e += 128) atomicAdd(&numacc[b * 192 + e], numl[e]);
}

// ---------------------------------------------------------------------------
// K4: palette[b][c][n] = num/(den+1e-8)
// ---------------------------------------------------------------------------
__global__ void k4_palette(const float* __restrict__ numacc,
                           const float* __restrict__ denacc,
                           float* __restrict__ pal) {
  const int t = blockIdx.x * blockDim.x + threadIdx.x;
  if (t < 1536) {
    const int b = t / 192, r = t % 192, n = r & 63;
    pal[t] = numacc[t] / (denacc[b * 64 + n] + 1e-8f);
  }
}

// ---------------------------------------------------------------------------
// K5: transformed[b][c][hw] = sum_k w_k * palette[b][c][idx_k]
// ---------------------------------------------------------------------------
__global__ __launch_bounds__(256) void k5_recon(const unsigned int* __restrict__ tkidx,
                                                const float* __restrict__ tkw,
                                                const float* __restrict__ pal,
                                                float* __restrict__ out) {
  __shared__ float pl[192];
  const int p = blockIdx.x * 256 + threadIdx.x;
  const int b = p >> 16;
  if (threadIdx.x < 192) pl[threadIdx.x] = pal[b * 192 + threadIdx.x];
  __syncthreads();
  const unsigned pk = tkidx[p];
  const v4f w = ((const v4f*)tkw)[p];
  const int i0 = pk & 255, i1 = (pk >> 8) & 255, i2 = (pk >> 16) & 255, i3 = pk >> 24;
  const int hw = p & (HW - 1);
  float* o = out + (size_t)b * 3 * HW + hw;
#pragma unroll
  for (int c = 0; c < 3; ++c) {
    const float* pc = &pl[c * 64];
    o[(size_t)c * HW] = w.x * pc[i0] + w.y * pc[i1] + w.z * pc[i2] + w.w * pc[i3];
  }
}

// ---------------------------------------------------------------------------
extern "C" void kernel_launch(void* const* d_in, const int* in_sizes, int n_in,
                              void* d_out, int out_size, void* d_ws, size_t ws_size,
                              hipStream_t stream) {
  (void)in_sizes; (void)n_in; (void)out_size; (void)ws_size;
  const float* img   = (const float*)d_in[0];
  const float* feat  = (const float*)d_in[1];
  const float* wb    = (const float*)d_in[2];
  const float* bb    = (const float*)d_in[3];
  const float* gamma = (const float*)d_in[4];
  const float* beta  = (const float*)d_in[5];
  const float* wmask = (const float*)d_in[6];

  float* out   = (float*)d_out;
  float* trans = out;                          // [8,3,256,256]  = 1,572,864
  float* m_out = out + 1572864;                // [8,64,256,256] = 33,554,432
  float* pal   = out + 1572864 + 33554432;     // [8,3,64]       = 1,536

  char* ws = (char*)d_ws;
  float* xws    = (float*)ws;                              // 524288*16 f32
  float* stats  = (float*)(ws + (size_t)33554432);         // 32 f32 (sum, sumsq)
  float* numacc = stats + 32;                              // 1536 f32
  float* denacc = numacc + 1536;                           // 512 f32
  unsigned int* tkidx = (unsigned int*)(denacc + 512);     // 524288 u32
  float* tkw    = (float*)(tkidx + 524288);                // 524288*4 f32 (16B aligned)

  hipMemsetAsync(stats, 0, (32 + 1536 + 512) * sizeof(float), stream);
  hipMemsetAsync(m_out, 0, (size_t)33554432 * sizeof(float), stream);

  k1_bneck<<<1024, 256, 0, stream>>>(feat, wb, bb, xws, stats);
  k3_main<<<512, 128, 0, stream>>>(xws, wmask, gamma, beta, stats, img,
                                   m_out, numacc, denacc, tkidx, tkw);
  k4_palette<<<6, 256, 0, stream>>>(numacc, denacc, pal);
  k5_recon<<<2048, 256, 0, stream>>>(tkidx, tkw, pal, trans);
}